// GCN_Yelp_1_13606456394531
// MI455X (gfx1250) — compile-verified
//
#include <hip/hip_runtime.h>
#include <math.h>

#define IN_DIM  300
#define OUT_DIM 100
#define OUT_PAD 112            // 7 x 16 column tiles, padded
#define KCHUNK  60             // K staging chunk (30 K-pairs)
#define PAIR_STRIDE (2 * OUT_PAD)   // floats per K-pair row in LDS (224)

typedef __attribute__((ext_vector_type(2))) float v2f;
typedef __attribute__((ext_vector_type(8))) float v8f;

// ---------------- degree = 1 (self loop) ----------------
__global__ void k_deg_init(float* __restrict__ deg, int n) {
  int i = blockIdx.x * blockDim.x + threadIdx.x;
  if (i < n) deg[i] = 1.0f;
}

// ---------------- deg[dst] += 1 per edge ----------------
__global__ void k_deg_scatter(const int* __restrict__ dst, float* __restrict__ deg, int e) {
  int i = blockIdx.x * blockDim.x + threadIdx.x;
  if (i < e) {
    (void)__hip_atomic_fetch_add(&deg[dst[i]], 1.0f,
                                 __ATOMIC_RELAXED, __HIP_MEMORY_SCOPE_AGENT);
  }
}

// ---------------- deg -> deg^{-1/2} in place ----------------
__global__ void k_dinv(float* __restrict__ deg, int n) {
  int i = blockIdx.x * blockDim.x + threadIdx.x;
  if (i < n) deg[i] = rsqrtf(deg[i]);   // deg >= 1 always (self loops)
}

// ---------------- h = x @ W via V_WMMA_F32_16X16X4_F32 ----------------
// Block: 256 threads = 8 waves; each wave owns a 16-row stripe, all 7 col tiles.
// LDS layout: K-pair interleaved, Ws[(k/2)*224 + c*2 + (k&1)], so a lane's
// B fragment {W[k][c], W[k+1][c]} is one contiguous 8-byte ds_load_b64.
__global__ __launch_bounds__(256)
void k_gemm_wmma(const float* __restrict__ x, const float* __restrict__ W,
                 float* __restrict__ h, int n) {
  extern __shared__ float Ws[];                 // [KCHUNK/2][PAIR_STRIDE]
  const int tid  = threadIdx.x;
  const int lane = tid & 31;
  const int wave = tid >> 5;
  const int m0   = (blockIdx.x * 8 + wave) * 16;
  const int half = lane >> 4;                   // 0: lanes 0-15, 1: lanes 16-31
  const int l15  = lane & 15;
  const int kb   = half * 2;                    // K sub-pair per lane half

  v8f acc[7];
  v8f z = {};
#pragma unroll
  for (int t = 0; t < 7; ++t) acc[t] = z;

  // A fragment source row (clamped for safety; n % 16 == 0 in practice)
  int arow = m0 + l15; if (arow >= n) arow = n - 1;
  const float* xrow = x + (size_t)arow * IN_DIM;

  for (int kc = 0; kc < IN_DIM; kc += KCHUNK) {
    __syncthreads();
    // Stage W[kc..kc+KCHUNK-1, :] into LDS, pair-interleaved, zero-padded cols
    for (int idx = tid; idx < (KCHUNK / 2) * PAIR_STRIDE; idx += 256) {
      int pair = idx / PAIR_STRIDE;
      int rem  = idx - pair * PAIR_STRIDE;
      int c    = rem >> 1;
      int sub  = rem & 1;
      int k    = kc + pair * 2 + sub;
      Ws[idx] = (c < OUT_DIM) ? W[k * OUT_DIM + c] : 0.0f;
    }
    __syncthreads();

    if (m0 < n) {
#pragma unroll 5
      for (int k0 = 0; k0 < KCHUNK; k0 += 4) {
        // A: 16x4 f32; lane half selects K pair {kb, kb+1}; 8-byte aligned load
        v2f a = *(const v2f*)(xrow + kc + k0 + kb);
        const float* bbase = Ws + ((k0 >> 1) + half) * PAIR_STRIDE;
#pragma unroll
        for (int t = 0; t < 7; ++t) {
          // Contiguous 8-byte LDS load of {W[k][col], W[k+1][col]}
          v2f bf = *(const v2f*)(bbase + (t * 16 + l15) * 2);
          acc[t] = __builtin_amdgcn_wmma_f32_16x16x4_f32(
              /*neg_a=*/false, a, /*neg_b=*/false, bf,
              /*c_mod=*/(short)0, acc[t], /*reuse_a=*/false, /*reuse_b=*/false);
        }
      }
    }
  }

  if (m0 < n) {
    const int mbase = m0 + half * 8;            // C/D layout: VGPR g -> M = g + 8*half
#pragma unroll
    for (int t = 0; t < 7; ++t) {
      int col = t * 16 + l15;
      if (col < OUT_DIM) {
#pragma unroll
        for (int g = 0; g < 8; ++g) {
          int m = mbase + g;
          if (m < n) h[(size_t)m * OUT_DIM + col] = acc[t][g];
        }
      }
    }
  }
}

// ---------------- out = b + h * dinv^2 (self-loop term; also inits d_out) ----------------
__global__ void k_out_init(const float* __restrict__ h, const float* __restrict__ dinv,
                           const float* __restrict__ b, float* __restrict__ out, int n) {
  int idx = blockIdx.x * blockDim.x + threadIdx.x;
  int total = n * OUT_DIM;
  if (idx < total) {
    int i = idx / OUT_DIM;
    int c = idx - i * OUT_DIM;
    float di = dinv[i];
    out[idx] = b[c] + h[idx] * di * di;
  }
}

// ---------------- edge scatter: one wave32 per edge ----------------
__global__ void k_edge_scatter(const int* __restrict__ src, const int* __restrict__ dst,
                               const float* __restrict__ h, const float* __restrict__ dinv,
                               float* __restrict__ out, int e) {
  int wid = blockIdx.x * (blockDim.x >> 5) + (threadIdx.x >> 5);
  if (wid >= e) return;
  int lane = threadIdx.x & 31;
  int s = src[wid];
  int d = dst[wid];
  float w = dinv[s] * dinv[d];
  const float* hr   = h   + (size_t)s * OUT_DIM;
  float*       orow = out + (size_t)d * OUT_DIM;
  for (int c = lane; c < OUT_DIM; c += 32) {
    (void)__hip_atomic_fetch_add(&orow[c], hr[c] * w,
                                 __ATOMIC_RELAXED, __HIP_MEMORY_SCOPE_AGENT);
  }
}

extern "C" void kernel_launch(void* const* d_in, const int* in_sizes, int n_in,
                              void* d_out, int out_size, void* d_ws, size_t ws_size,
                              hipStream_t stream) {
  const float* x  = (const float*)d_in[0];
  const int*   ei = (const int*)d_in[1];
  const float* W  = (const float*)d_in[2];
  const float* b  = (const float*)d_in[3];
  float*       out = (float*)d_out;

  const int n = in_sizes[0] / IN_DIM;   // 100000
  const int e = in_sizes[1] / 2;        // 3200000
  const int* src = ei;                  // edge_index[0]
  const int* dst = ei + e;              // edge_index[1]

  // Workspace: h [n*OUT_DIM] floats, then deg/dinv [n] floats (~40.4 MB)
  float* h   = (float*)d_ws;
  float* deg = h + (size_t)n * OUT_DIM;

  k_deg_init<<<(n + 255) / 256, 256, 0, stream>>>(deg, n);
  k_deg_scatter<<<(e + 255) / 256, 256, 0, stream>>>(dst, deg, e);
  k_dinv<<<(n + 255) / 256, 256, 0, stream>>>(deg, n);

  const int rows_per_block = 8 * 16;    // 8 waves x 16 rows
  const int mblocks = (n + rows_per_block - 1) / rows_per_block;
  k_gemm_wmma<<<mblocks, 256, (KCHUNK / 2) * PAIR_STRIDE * sizeof(float), stream>>>(x, W, h, n);

  const int total = n * OUT_DIM;
  k_out_init<<<(total + 255) / 256, 256, 0, stream>>>(h, deg, b, out, n);

  k_edge_scatter<<<(e + 7) / 8, 256, 0, stream>>>(src, dst, h, deg, out, e);
}